// Resample2d_34522947125558
// MI455X (gfx1250) — compile-verified
//
#include <hip/hip_runtime.h>
#include <math.h>

#define BB 8
#define CC 64
#define HH 256
#define WW 256
#define HWSZ (HH * WW)

typedef float v4f __attribute__((ext_vector_type(4)));

__global__ __launch_bounds__(256, 2) void resample2d_fast(
    const float* __restrict__ img,
    const float* __restrict__ flow,
    const int* __restrict__ p_dil,
    const int* __restrict__ p_sig,
    float* __restrict__ out)
{
    const int w = threadIdx.x;                 // 0..255 (pixel x)
    const int h = blockIdx.x & (HH - 1);       // pixel y
    const int b = blockIdx.x >> 8;             // batch

    __shared__ float sflow[2 * WW];

    // --- CDNA5 async global->LDS staging of the flow row (ASYNCcnt path) ---
    const float* gx = flow + (size_t)(b * 2 + 0) * HWSZ + h * WW + w;
    const float* gy = flow + (size_t)(b * 2 + 1) * HWSZ + h * WW + w;
    // low 32 bits of a generic pointer into __shared__ == LDS byte offset
    unsigned l0 = (unsigned)(size_t)&sflow[w];
    unsigned l1 = (unsigned)(size_t)&sflow[WW + w];
    asm volatile("global_load_async_to_lds_b32 %0, %1, off"
                 :: "v"(l0), "v"(gx) : "memory");
    asm volatile("global_load_async_to_lds_b32 %0, %1, off"
                 :: "v"(l1), "v"(gy) : "memory");
    asm volatile("s_wait_asynccnt 0" ::: "memory");
    __syncthreads();

    const float fx = sflow[w];
    const float fy = sflow[WW + w];

    const int   dil    = p_dil[0];
    const float sig    = (float)p_sig[0];
    const float inv2s2 = 1.0f / (2.0f * sig * sig);

    const float xf = (float)w + fx;
    const float yf = (float)h + fy;
    const float x0 = __builtin_floorf(xf);
    const float y0 = __builtin_floorf(yf);

    // 4 separable x-weights / y-weights + clamped integer taps (once per pixel)
    float wx[4], wy[4];
    int xi[4], yi[4];
#pragma unroll
    for (int t = 0; t < 4; ++t) {
        const float xs = x0 + (float)(t * dil);
        const float ys = y0 + (float)(t * dil);
        const float dx = xs - xf;
        const float dy = ys - yf;
        wx[t] = expf(-dx * dx * inv2s2);
        wy[t] = expf(-dy * dy * inv2s2);
        xi[t] = min(max((int)xs, 0), WW - 1);   // trunc-then-clamp, like reference
        yi[t] = min(max((int)ys, 0), HH - 1);
    }
    // wsum = (sum wy_i) * (sum wx_j); fold 1/(wsum+1e-8) into the weights
    const float inv = 1.0f /
        ((wx[0] + wx[1] + wx[2] + wx[3]) * (wy[0] + wy[1] + wy[2] + wy[3]) + 1e-8f);

    float cw[4][4];
#pragma unroll
    for (int i = 0; i < 4; ++i)
#pragma unroll
        for (int j = 0; j < 4; ++j)
            cw[i][j] = wy[i] * wx[j] * inv;

    const float* __restrict__ base  = img + (size_t)b * CC * HWSZ;
    float* __restrict__       obase = out + (size_t)b * CC * HWSZ + h * WW + w;

    const int  ix0  = (int)x0;
    const bool fast = (dil == 1) & (ix0 >= 0) & (ix0 + 3 < WW);

    if (fast) {
        // dilation==1 interior: 4 consecutive x-taps -> one b128 gather per row
        int ro[4];
#pragma unroll
        for (int i = 0; i < 4; ++i) ro[i] = yi[i] * WW + ix0;

#pragma unroll 4
        for (int c = 0; c < CC; ++c) {
            const float* p = base + (size_t)c * HWSZ;
            float acc = 0.0f;
#pragma unroll
            for (int i = 0; i < 4; ++i) {
                v4f s = *(const v4f*)(p + ro[i]);   // global_load_b128 gather (L2-cached)
                acc += s.x * cw[i][0] + s.y * cw[i][1] +
                       s.z * cw[i][2] + s.w * cw[i][3];
            }
            // write-once output: non-temporal so it doesn't evict input1 from L2
            __builtin_nontemporal_store(acc, &obase[(size_t)c * HWSZ]);
        }
    } else {
        // border / large-flow / generic-dilation path: fully clamped scalar gathers
        int idx[4][4];
#pragma unroll
        for (int i = 0; i < 4; ++i)
#pragma unroll
            for (int j = 0; j < 4; ++j)
                idx[i][j] = yi[i] * WW + xi[j];

#pragma unroll 2
        for (int c = 0; c < CC; ++c) {
            const float* p = base + (size_t)c * HWSZ;
            float acc = 0.0f;
#pragma unroll
            for (int i = 0; i < 4; ++i)
#pragma unroll
                for (int j = 0; j < 4; ++j)
                    acc += p[idx[i][j]] * cw[i][j];
            __builtin_nontemporal_store(acc, &obase[(size_t)c * HWSZ]);
        }
    }
}

extern "C" void kernel_launch(void* const* d_in, const int* in_sizes, int n_in,
                              void* d_out, int out_size, void* d_ws, size_t ws_size,
                              hipStream_t stream) {
    // dict order: input1, input2, kernel_size, dilation, sigma
    const float* img  = (const float*)d_in[0];
    const float* flow = (const float*)d_in[1];
    const int*   dil  = (const int*)d_in[3];
    const int*   sig  = (const int*)d_in[4];
    float*       out  = (float*)d_out;

    dim3 grid(BB * HH);   // one block per (b, h) row
    dim3 block(WW);       // 256 threads = 8 wave32s, thread = pixel
    hipLaunchKernelGGL(resample2d_fast, grid, block, 0, stream,
                       img, flow, dil, sig, out);
}